// GIN_75935021794036
// MI455X (gfx1250) — compile-verified
//
#include <hip/hip_runtime.h>
#include <math.h>

#define HID 128
#define ROWS_PER_BLOCK 64   // 4 x 16-row WMMA tiles share each B fragment

typedef __attribute__((ext_vector_type(2))) float v2f;
typedef __attribute__((ext_vector_type(8))) float v8f;

// ---------------------------------------------------------------------------
// float atomic max via monotonic int mapping (init must be -inf)
// ---------------------------------------------------------------------------
__device__ __forceinline__ void atomicMaxFloat(float* addr, float value) {
  if (value >= 0.0f)
    atomicMax((int*)addr, __float_as_int(value));
  else
    atomicMin((unsigned int*)addr, __float_as_uint(value));
}

// ---------------------------------------------------------------------------
// h = x (identity term of GIN aggregation), float4-vectorized copy
// ---------------------------------------------------------------------------
__global__ void copy4_kernel(const float* __restrict__ src,
                             float* __restrict__ dst, int n4) {
  int i = blockIdx.x * blockDim.x + threadIdx.x;
  if (i < n4) ((float4*)dst)[i] = ((const float4*)src)[i];
}

// ---------------------------------------------------------------------------
// h[dst[e]] += x[src[e]] : edge-parallel scatter-add.
// One thread = (edge, 4-feature chunk): float4 gather + 4 f32 atomics.
// Working set (2 x 25.6 MB node buffers) is L2-resident on the 192 MB L2.
// ---------------------------------------------------------------------------
__global__ void scatter_add_kernel(const float* __restrict__ x,
                                   const int* __restrict__ src,
                                   const int* __restrict__ dst,
                                   float* h, int nedges, int din) {
  const int chunks = din >> 2;
  int tid = blockIdx.x * blockDim.x + threadIdx.x;
  if (tid >= nedges * chunks) return;
  int e = tid / chunks;
  int c = (tid - e * chunks) << 2;
  int s = src[e];
  int d = dst[e];
  const float4 v = *(const float4*)(x + (size_t)s * din + c);
  float* p = h + (size_t)d * din + c;
  atomicAdd(p + 0, v.x);
  atomicAdd(p + 1, v.y);
  atomicAdd(p + 2, v.z);
  atomicAdd(p + 3, v.w);
}

// ---------------------------------------------------------------------------
// Fused GIN MLP: out = elu(h @ W1 + b1) @ W2 + b2 for a 64-row node tile.
// 256 threads = 8 wave32s; wave w owns output columns [16w, 16w+16) and
// computes 4 stacked 16x16 output tiles, so each B fragment (weight slab)
// is loaded once per K-step and reused by 4 WMMAs -> 4x less W traffic.
// Exact fp32 math via V_WMMA_F32_16X16X4_F32.
//
// Fragment layouts (ISA 7.12.2, 32-bit, 16x4 / 4x16 / 16x16):
//   A: lane L -> M = L&15 ; a.x = A[M][k0+2*(L>=16)], a.y = A[M][k0+2*(L>=16)+1]
//   B: lane L -> N = L&15 ; b.x = B[k0+2*(L>=16)][N], b.y = B[k0+2*(L>=16)+1][N]
//   C/D: VGPR r, lane L   -> row = r + 8*(L>=16), col = L&15
//
// Safe for out==h (in-place): a block reads only its own 64 rows into LDS
// (before __syncthreads) and writes only those same rows afterwards.
// ---------------------------------------------------------------------------
__global__ void __launch_bounds__(256) gin_mlp_kernel(
    const float* h, const float* __restrict__ W1, const float* __restrict__ b1,
    const float* __restrict__ W2, const float* __restrict__ b2,
    float* out, int n, int din) {
  __shared__ float sh[ROWS_PER_BLOCK * HID];  // input tile  [64 x din]
  __shared__ float st[ROWS_PER_BLOCK * HID];  // hidden tile [64 x HID]

  const int tid  = threadIdx.x;
  const int wave = tid >> 5;
  const int lane = tid & 31;
  const int lh   = lane & 15;    // M for A-frag / N for B-frag
  const int khi  = lane >> 4;    // K sub-offset selector (0 or 1)
  const int col0 = wave << 4;    // 16 output cols per wave
  const int rowBase = blockIdx.x * ROWS_PER_BLOCK;

  // stage the 64 x din input tile into LDS (zero-fill past n)
  for (int idx = tid; idx < ROWS_PER_BLOCK * din; idx += 256) {
    int r = idx / din;
    int c = idx - r * din;
    int gr = rowBase + r;
    sh[r * din + c] = (gr < n) ? h[(size_t)gr * din + c] : 0.0f;
  }
  __syncthreads();

  // ---- stage 1: hidden = elu(h @ W1 + b1) -------------------------------
  v8f acc[4] = {{}, {}, {}, {}};
  for (int k0 = 0; k0 < din; k0 += 4) {
    const int ka = k0 + 2 * khi;
    v2f b;
    b.x = W1[ka * HID + col0 + lh];
    b.y = W1[(ka + 1) * HID + col0 + lh];
#pragma unroll
    for (int t = 0; t < 4; ++t) {
      v2f a;
      a.x = sh[(t * 16 + lh) * din + ka];
      a.y = sh[(t * 16 + lh) * din + ka + 1];
      acc[t] = __builtin_amdgcn_wmma_f32_16x16x4_f32(
          /*neg_a=*/false, a, /*neg_b=*/false, b,
          /*c_mod=*/(short)0, acc[t], /*reuse_a=*/false, /*reuse_b=*/false);
    }
  }
  const float bias1 = b1[col0 + lh];
#pragma unroll
  for (int t = 0; t < 4; ++t) {
#pragma unroll
    for (int r = 0; r < 8; ++r) {
      const int row = t * 16 + r + 8 * khi;
      float v = acc[t][r] + bias1;
      v = (v > 0.0f) ? v : (__expf(v) - 1.0f);  // ELU, alpha = 1
      st[row * HID + col0 + lh] = v;
    }
  }
  __syncthreads();

  // ---- stage 2: out = hidden @ W2 + b2 ----------------------------------
  v8f acc2[4] = {{}, {}, {}, {}};
  for (int k0 = 0; k0 < HID; k0 += 4) {
    const int ka = k0 + 2 * khi;
    v2f b;
    b.x = W2[ka * HID + col0 + lh];
    b.y = W2[(ka + 1) * HID + col0 + lh];
#pragma unroll
    for (int t = 0; t < 4; ++t) {
      v2f a;
      a.x = st[(t * 16 + lh) * HID + ka];
      a.y = st[(t * 16 + lh) * HID + ka + 1];
      acc2[t] = __builtin_amdgcn_wmma_f32_16x16x4_f32(
          false, a, false, b, (short)0, acc2[t], false, false);
    }
  }
  const float bias2 = b2[col0 + lh];
#pragma unroll
  for (int t = 0; t < 4; ++t) {
#pragma unroll
    for (int r = 0; r < 8; ++r) {
      const int gr = rowBase + t * 16 + r + 8 * khi;
      if (gr < n) out[(size_t)gr * HID + col0 + lh] = acc2[t][r] + bias2;
    }
  }
}

// ---------------------------------------------------------------------------
// d_out init: [256 graphs][256]: cols 0..127 (sum pool) = 0,
//                                cols 128..255 (max pool) = -inf
// ---------------------------------------------------------------------------
__global__ void init_out_kernel(float* out, int total) {
  int i = blockIdx.x * blockDim.x + threadIdx.x;
  if (i >= total) return;
  int j = i & (2 * HID - 1);
  out[i] = (j < HID) ? 0.0f : __int_as_float(0xff800000);
}

// ---------------------------------------------------------------------------
// Readout: gate = sigmoid(x . Wg + bg); sum-pool(gate*x) || max-pool(x).
// One wave32 per node; wave-level shuffle reduction for the gate dot.
// ---------------------------------------------------------------------------
__global__ void __launch_bounds__(256) readout_kernel(
    const float* __restrict__ x, const int* __restrict__ batch,
    const float* __restrict__ Wg, const float* __restrict__ bg,
    float* out, int n) {
  const int node = blockIdx.x * 8 + (threadIdx.x >> 5);
  const int lane = threadIdx.x & 31;
  if (node >= n) return;
  const float* xi = x + (size_t)node * HID;

  float partial = 0.0f;
#pragma unroll
  for (int j = 0; j < HID; j += 32) partial += xi[j + lane] * Wg[j + lane];
#pragma unroll
  for (int off = 16; off > 0; off >>= 1)
    partial += __shfl_xor(partial, off, 32);

  const float gate = 1.0f / (1.0f + __expf(-(partial + bg[0])));
  const int g = batch[node];
  float* po = out + (size_t)g * 2 * HID;
#pragma unroll
  for (int j = 0; j < HID; j += 32) {
    const float v = xi[j + lane];
    atomicAdd(po + j + lane, gate * v);
    atomicMaxFloat(po + HID + j + lane, v);
  }
}

// ---------------------------------------------------------------------------
extern "C" void kernel_launch(void* const* d_in, const int* in_sizes, int n_in,
                              void* d_out, int out_size, void* d_ws,
                              size_t ws_size, hipStream_t stream) {
  (void)n_in; (void)ws_size;
  const float* x     = (const float*)d_in[0];
  const int*   ei    = (const int*)d_in[1];
  const int*   batch = (const int*)d_in[2];
  const float* W1[3] = {(const float*)d_in[3], (const float*)d_in[7],  (const float*)d_in[11]};
  const float* b1[3] = {(const float*)d_in[4], (const float*)d_in[8],  (const float*)d_in[12]};
  const float* W2[3] = {(const float*)d_in[5], (const float*)d_in[9],  (const float*)d_in[13]};
  const float* b2[3] = {(const float*)d_in[6], (const float*)d_in[10], (const float*)d_in[14]};
  const float* Wg = (const float*)d_in[15];
  const float* bg = (const float*)d_in[16];
  float* out = (float*)d_out;

  const int n      = in_sizes[2];       // 50000 nodes
  const int nedges = in_sizes[1] / 2;   // 800000 edges
  const int* src = ei;                  // edge_index[0]
  const int* dst = ei + nedges;         // edge_index[1]

  // workspace: two N x HID fp32 ping-pong buffers (2 * 25.6 MB)
  float* bufA = (float*)d_ws;
  float* bufB = bufA + (size_t)n * HID;

  const int B = 256;
  auto cdiv = [](int a, int b) { return (a + b - 1) / b; };

  // ---- layer 0 (din = 64): h = bufA, x_out = bufB -----------------------
  copy4_kernel<<<cdiv(n * 64 / 4, B), B, 0, stream>>>(x, bufA, n * 64 / 4);
  scatter_add_kernel<<<cdiv(nedges * (64 / 4), B), B, 0, stream>>>(
      x, src, dst, bufA, nedges, 64);
  gin_mlp_kernel<<<cdiv(n, ROWS_PER_BLOCK), B, 0, stream>>>(
      bufA, W1[0], b1[0], W2[0], b2[0], bufB, n, 64);

  // ---- layer 1 (din = 128): h = bufA, x_out = bufA (in-place MLP) -------
  copy4_kernel<<<cdiv(n * HID / 4, B), B, 0, stream>>>(bufB, bufA, n * HID / 4);
  scatter_add_kernel<<<cdiv(nedges * (HID / 4), B), B, 0, stream>>>(
      bufB, src, dst, bufA, nedges, HID);
  gin_mlp_kernel<<<cdiv(n, ROWS_PER_BLOCK), B, 0, stream>>>(
      bufA, W1[1], b1[1], W2[1], b2[1], bufA, n, HID);

  // ---- layer 2 (din = 128): h = bufB, x_out = bufB ----------------------
  copy4_kernel<<<cdiv(n * HID / 4, B), B, 0, stream>>>(bufA, bufB, n * HID / 4);
  scatter_add_kernel<<<cdiv(nedges * (HID / 4), B), B, 0, stream>>>(
      bufA, src, dst, bufB, nedges, HID);
  gin_mlp_kernel<<<cdiv(n, ROWS_PER_BLOCK), B, 0, stream>>>(
      bufB, W1[2], b1[2], W2[2], b2[2], bufB, n, HID);

  // ---- readout ----------------------------------------------------------
  init_out_kernel<<<cdiv(out_size, B), B, 0, stream>>>(out, out_size);
  readout_kernel<<<cdiv(n, 8), B, 0, stream>>>(bufB, batch, Wg, bg, out, n);
}